// SoftNDCGLoss_18614388261531
// MI455X (gfx1250) — compile-verified
//
#include <hip/hip_runtime.h>
#include <hip/hip_bf16.h>

// SoftNDCG loss, N = 2048.
//
// Strategy (MI455X): the 2048x2048 Sinkhorn matrix is never materialized.
// Every sinkhorn norm() step is exactly a diagonal rescale, so we keep
//   m = diag(r) * E * diag(c),  E_ij = exp(-|ss_i - y_j|)
// and recompute E on the fly. Each half-iteration becomes a weighted
// mat-vec against the implicit E, reading only ~24KB of vectors (LDS
// resident) instead of sweeping a 16MB matrix 200 times -> compute bound
// (4.2M exp+MAC per pass) instead of bandwidth bound.
//
// The per-tile reduction uses V_WMMA_F32_16X16X4_F32 (full f32 precision):
// A(16x4) carries the weight vector replicated across rows, B(4x16) the
// freshly computed exp() tile, D accumulates column sums.

#define SN_N     2048
#define SN_TOL   1e-6f
#define SN_EPS   1e-10f

typedef float v2f __attribute__((ext_vector_type(2)));
typedef float v8f __attribute__((ext_vector_type(8)));

#if __has_builtin(__builtin_amdgcn_wmma_f32_16x16x4_f32)
#define SN_HAVE_WMMA4 1
#else
#define SN_HAVE_WMMA4 0
#endif

// ---------------------------------------------------------------------------
// Bitonic sort (descending) of 2048 floats in LDS, 1024 threads.
// ---------------------------------------------------------------------------
__device__ __forceinline__ void bitonic_desc_2048(float* buf, unsigned t) {
  for (unsigned size = 2; size <= 2048u; size <<= 1) {
    for (unsigned stride = size >> 1; stride > 0; stride >>= 1) {
      __syncthreads();
      unsigned i = (t / stride) * (stride << 1) + (t % stride);
      unsigned j = i + stride;
      float a = buf[i], b = buf[j];
      bool up = ((i & size) == 0);           // "up" segments sorted descending
      if (up ? (a < b) : (a > b)) { buf[i] = b; buf[j] = a; }
    }
  }
  __syncthreads();
}

// ---------------------------------------------------------------------------
// prep: block0 sorts y_pred (descending) -> ss.
//       block1 computes labels = 2^y_true - 1, sorts a copy for IDCG,
//       reduces IDCG, resets c / ones / flags / devrow for this call.
// ---------------------------------------------------------------------------
__global__ __launch_bounds__(1024) void sn_prep(const float* __restrict__ y_pred,
                                                const float* __restrict__ y_true,
                                                float* __restrict__ ss,
                                                float* __restrict__ labels,
                                                float* __restrict__ c,
                                                float* __restrict__ ones,
                                                float* __restrict__ scal, // [0]=idcg [1]=devrow
                                                int* __restrict__ flags)  // [0]=done [1]=always-zero
{
  __shared__ float buf[2048];
  __shared__ float red[1024];
  const unsigned t = threadIdx.x;

  if (blockIdx.x == 0) {
    buf[t]        = y_pred[t];
    buf[t + 1024] = y_pred[t + 1024];
    bitonic_desc_2048(buf, t);
    ss[t]        = buf[t];
    ss[t + 1024] = buf[t + 1024];
  } else {
    float l0 = exp2f(y_true[t])        - 1.0f;
    float l1 = exp2f(y_true[t + 1024]) - 1.0f;
    labels[t]        = l0;
    labels[t + 1024] = l1;
    buf[t]        = l0;
    buf[t + 1024] = l1;
    bitonic_desc_2048(buf, t);
    // IDCG = sum(sorted_labels / log2(i+2))
    float v = buf[t] / log2f((float)(t + 2)) +
              buf[t + 1024] / log2f((float)(t + 1026));
    red[t] = v;
    __syncthreads();
    for (int s = 512; s > 0; s >>= 1) {
      if ((int)t < s) red[t] += red[t + s];
      __syncthreads();
    }
    c[t] = 1.0f;  c[t + 1024] = 1.0f;
    ones[t] = 1.0f; ones[t + 1024] = 1.0f;
    if (t == 0) {
      scal[0]  = red[0];   // idcg
      scal[1]  = 1e30f;    // devrow (blocked until first real row update)
      flags[0] = 0;        // done
      flags[1] = 0;        // permanent zero (forces non-gated passes)
    }
  }
}

// ---------------------------------------------------------------------------
// Implicit weighted mat-vec:  out_n = sum_k w[k]*w2[k] * exp(-|rowvals_k - colvals_n|)
// Grid (16, 8), block 256 (8 waves). Wave handles a 16-column tile over a
// 256-row chunk; per 4-row step one V_WMMA_F32_16X16X4_F32 accumulates the
// weighted column sums in fp32. Partial results per row-chunk go to
// partials[chunk*2048 + n]; a single-block kernel folds the 8 partials.
// ---------------------------------------------------------------------------
__global__ __launch_bounds__(256) void sn_pass(const float* __restrict__ rowvals,
                                               const float* __restrict__ colvals,
                                               const float* __restrict__ w,
                                               const float* __restrict__ w2,
                                               const int* __restrict__ done,
                                               float* __restrict__ partials)
{
  __shared__ __align__(16) float s_row[256];
  __shared__ __align__(16) float s_w[256];

  if (*done) return;                       // uniform across block

  const int tid = threadIdx.x;
  const int rowbase = blockIdx.y * 256;
  s_row[tid] = rowvals[rowbase + tid];
  s_w[tid]   = w[rowbase + tid] * w2[rowbase + tid];
  __syncthreads();

  const int wave = tid >> 5;
  const int lane = tid & 31;
  const int colbase = (blockIdx.x * 8 + wave) * 16;
  const float cv = colvals[colbase + (lane & 15)];
  const int khalf = (lane >> 4) << 1;      // lanes 0-15 -> k%4 in {0,1}; 16-31 -> {2,3}

#if SN_HAVE_WMMA4
  v8f acc;
  #pragma unroll
  for (int q = 0; q < 8; ++q) acc[q] = 0.0f;
#else
  float facc = 0.0f;
#endif

  for (int m4 = 0; m4 < 64; ++m4) {
    const int kb = (m4 << 2) + khalf;      // even -> 8B aligned in LDS
    v2f a = *(const v2f*)&s_w[kb];
    v2f s = *(const v2f*)&s_row[kb];
    v2f b;
    b.x = __expf(-fabsf(s.x - cv));
    b.y = __expf(-fabsf(s.y - cv));
#if SN_HAVE_WMMA4
    // A[m,k] = weight_k (replicated over m), B[k,n] = E_kn  ->  D[*,n] = sum_k w_k E_kn
    acc = __builtin_amdgcn_wmma_f32_16x16x4_f32(
        /*neg_a=*/false, a, /*neg_b=*/false, b,
        /*c_mod=*/(short)0, acc, /*reuse_a=*/false, /*reuse_b=*/false);
#else
    facc += a.x * b.x + a.y * b.y;
#endif
  }

#if SN_HAVE_WMMA4
  // D row M=0 lives in vgpr0 of lanes 0..15 (all rows identical anyway).
  if (lane < 16) partials[blockIdx.y * SN_N + colbase + lane] = acc[0];
#else
  float v = facc;
  v += __shfl_down(v, 16);                 // combine k-halves (wave32)
  if (lane < 16) partials[blockIdx.y * SN_N + colbase + lane] = v;
#endif
}

// ---------------------------------------------------------------------------
// r_i = 1 / Z_i   (initial row softmax folded into r; Z >= 1 since min|diff|=0)
// ---------------------------------------------------------------------------
__global__ __launch_bounds__(256) void sn_init_r(const float* __restrict__ partials,
                                                 float* __restrict__ r)
{
  const int tid = threadIdx.x;
  for (int g = 0; g < 8; ++g) {
    const int i = tid + 256 * g;
    float z = 0.0f;
    for (int p = 0; p < 8; ++p) z += partials[p * SN_N + i];
    r[i] = 1.0f / z;
  }
}

// ---------------------------------------------------------------------------
// Column normalize + deferred convergence detection.
// colsum_j (of m from the *previous* iteration) = c_j * S_j.
// If allow && max|colsum-1| < TOL && stored devrow < TOL -> freeze (done=1).
// ---------------------------------------------------------------------------
__global__ __launch_bounds__(256) void sn_update_c(const float* __restrict__ partials,
                                                   float* __restrict__ c,
                                                   const float* __restrict__ scal, // [1]=devrow
                                                   int* __restrict__ done,
                                                   int allow)
{
  __shared__ float red[256];
  __shared__ int sdone;
  if (*done) return;

  const int tid = threadIdx.x;
  float colsum[8];
  float dev = 0.0f;
  for (int g = 0; g < 8; ++g) {
    const int j = tid + 256 * g;
    float S = 0.0f;
    for (int p = 0; p < 8; ++p) S += partials[p * SN_N + j];
    const float cs = c[j] * S;
    colsum[g] = cs;
    dev = fmaxf(dev, fabsf(cs - 1.0f));
  }
  red[tid] = dev;
  __syncthreads();
  for (int s = 128; s > 0; s >>= 1) {
    if (tid < s) red[tid] = fmaxf(red[tid], red[tid + s]);
    __syncthreads();
  }
  if (tid == 0) {
    int d = (allow && red[0] < SN_TOL && scal[1] < SN_TOL) ? 1 : 0;
    sdone = d;
    if (d) *done = 1;
  }
  __syncthreads();
  if (!sdone) {
    for (int g = 0; g < 8; ++g) {
      const int j = tid + 256 * g;
      c[j] = c[j] / fmaxf(colsum[g], SN_EPS);
    }
  }
}

// ---------------------------------------------------------------------------
// Row normalize; record devrow = max|rowsum(m_new) - 1| for deferred check.
// ---------------------------------------------------------------------------
__global__ __launch_bounds__(256) void sn_update_r(const float* __restrict__ partials,
                                                   float* __restrict__ r,
                                                   float* __restrict__ scal, // [1]=devrow
                                                   const int* __restrict__ done)
{
  __shared__ float red[256];
  if (*done) return;

  const int tid = threadIdx.x;
  float dev = 0.0f;
  for (int g = 0; g < 8; ++g) {
    const int i = tid + 256 * g;
    float T = 0.0f;
    for (int p = 0; p < 8; ++p) T += partials[p * SN_N + i];
    const float rold = r[i];
    const float rs = rold * T;
    const float rn = rold / fmaxf(rs, SN_EPS);
    r[i] = rn;
    dev = fmaxf(dev, fabsf(rn * T - 1.0f));
  }
  red[tid] = dev;
  __syncthreads();
  for (int s = 128; s > 0; s >>= 1) {
    if (tid < s) red[tid] = fmaxf(red[tid], red[tid + s]);
    __syncthreads();
  }
  if (tid == 0) scal[1] = red[0];
}

// ---------------------------------------------------------------------------
// loss = 1 - (sum_i r_i * G_i / log2(i+2)) / idcg,  G_i = sum_j E_ij c_j lab_j
// ---------------------------------------------------------------------------
__global__ __launch_bounds__(256) void sn_finalize(const float* __restrict__ partials,
                                                   const float* __restrict__ r,
                                                   const float* __restrict__ scal, // [0]=idcg
                                                   float* __restrict__ out)
{
  __shared__ float red[256];
  const int tid = threadIdx.x;
  float acc = 0.0f;
  for (int g = 0; g < 8; ++g) {
    const int i = tid + 256 * g;
    float G = 0.0f;
    for (int p = 0; p < 8; ++p) G += partials[p * SN_N + i];
    acc += r[i] * G / log2f((float)(i + 2));
  }
  red[tid] = acc;
  __syncthreads();
  for (int s = 128; s > 0; s >>= 1) {
    if (tid < s) red[tid] += red[tid + s];
    __syncthreads();
  }
  if (tid == 0) out[0] = 1.0f - red[0] / scal[0];
}

// ---------------------------------------------------------------------------
extern "C" void kernel_launch(void* const* d_in, const int* in_sizes, int n_in,
                              void* d_out, int out_size, void* d_ws, size_t ws_size,
                              hipStream_t stream) {
  (void)in_sizes; (void)n_in; (void)out_size; (void)ws_size;
  const float* y_pred = (const float*)d_in[0];
  const float* y_true = (const float*)d_in[1];

  float* fws      = (float*)d_ws;
  float* partials = fws;              // 8 * 2048
  float* ss       = fws + 16384;      // sorted scores, descending
  float* labels   = fws + 18432;      // 2^y_true - 1 (original order)
  float* r        = fws + 20480;      // row scale
  float* c        = fws + 22528;      // col scale
  float* ones     = fws + 24576;
  float* scal     = fws + 26624;      // [0]=idcg, [1]=devrow
  int*   flags    = (int*)(fws + 26632); // [0]=done, [1]=zero

  sn_prep<<<2, 1024, 0, stream>>>(y_pred, y_true, ss, labels, c, ones, scal, flags);

  dim3 pg(16, 8);
  // Z_i = sum_j exp(-|ss_i - y_j|)  ->  out_i with rowvals=y, colvals=ss, w=1
  sn_pass<<<pg, 256, 0, stream>>>(y_pred, ss, ones, ones, flags + 1, partials);
  sn_init_r<<<1, 256, 0, stream>>>(partials, r);

  for (int t = 1; t <= 50; ++t) {
    // S_j = sum_i r_i exp(-|ss_i - y_j|)
    sn_pass<<<pg, 256, 0, stream>>>(ss, y_pred, r, ones, flags, partials);
    sn_update_c<<<1, 256, 0, stream>>>(partials, c, scal, flags, (t >= 2) ? 1 : 0);
    // T_i = sum_j exp(-|ss_i - y_j|) c_j
    sn_pass<<<pg, 256, 0, stream>>>(y_pred, ss, c, ones, flags, partials);
    sn_update_r<<<1, 256, 0, stream>>>(partials, r, scal, flags);
  }

  // G_i = sum_j exp(-|ss_i - y_j|) * c_j * labels_j  (always runs)
  sn_pass<<<pg, 256, 0, stream>>>(y_pred, ss, c, labels, flags + 1, partials);
  sn_finalize<<<1, 256, 0, stream>>>(partials, r, scal, (float*)d_out);
}